// Model_15307263443708
// MI455X (gfx1250) — compile-verified
//
#include <hip/hip_runtime.h>
#include <hip/hip_bf16.h>

// CSR SpMV: y[row] = sum_k values[k] * x[cols[k]] over k in [row_ptrs[row], row_ptrs[row+1]).
//
// MI455X design notes:
//  - Memory-bound (AI ~0.25 flop/byte): floor = (values+cols ~205MB)/23.3TB/s ~ 8.8us.
//    No dense sub-blocks (random cols) -> WMMA/TDM do not apply; this is a pure
//    data-movement kernel and its ceiling is the HBM stream.
//  - One wave32 per row. Main loop unrolled x2: each trip a wave issues
//    4x global_load_b128 (values+cols, chunks c and c+32) then 8 independent x
//    gathers -> maximal loads-in-flight per wave before the FMA dependency.
//    For the 256-nnz rows here that is the WHOLE row in one trip.
//  - values/cols are streamed exactly once -> non-temporal loads (th:TH_LOAD_NT)
//    so the 400KB x vector stays resident in the 192MB L2 for the gathers (RT).
//  - Wave32 shuffle tree reduction; single NT store of y[row].

typedef float vfloat4 __attribute__((ext_vector_type(4)));
typedef int   vint4   __attribute__((ext_vector_type(4)));

__global__ __launch_bounds__(256) void spmv_csr_wave32_kernel(
    const float* __restrict__ values,
    const int*   __restrict__ cols,
    const int*   __restrict__ row_ptrs,
    const float* __restrict__ x,
    float*       __restrict__ y,
    int num_rows)
{
    const int lane          = threadIdx.x & 31;           // wave32 lane id
    const int wave_in_block = threadIdx.x >> 5;           // 8 waves / 256-thread block
    const int row           = blockIdx.x * (blockDim.x >> 5) + wave_in_block;
    if (row >= num_rows) return;

    // Wave-uniform row bounds: all lanes hit the same b64 -> one L2 request.
    const int start = row_ptrs[row];
    const int end   = row_ptrs[row + 1];
    const int len   = end - start;

    float sum = 0.0f;

    if ((start & 3) == 0) {
        // Vectorized path (row start 16B-aligned; always true for uniform 256-nnz CSR).
        const vfloat4* __restrict__ vv = reinterpret_cast<const vfloat4*>(values + start);
        const vint4*   __restrict__ cv = reinterpret_cast<const vint4*>(cols + start);
        const int nvec = len >> 2;                        // float4 chunks in this row

        // Unrolled x2: chunks c and c+32 per trip = 256 elements per wave-trip.
        // Issues all 4 NT b128 stream loads + 8 gathers before the first use.
        int c = lane;
        for (; c + 32 < nvec; c += 64) {
            vfloat4 v0 = __builtin_nontemporal_load(vv + c);        // b128 th:NT
            vint4   i0 = __builtin_nontemporal_load(cv + c);        // b128 th:NT
            vfloat4 v1 = __builtin_nontemporal_load(vv + c + 32);   // b128 th:NT
            vint4   i1 = __builtin_nontemporal_load(cv + c + 32);   // b128 th:NT
            float x00 = x[i0.x], x01 = x[i0.y], x02 = x[i0.z], x03 = x[i0.w];
            float x10 = x[i1.x], x11 = x[i1.y], x12 = x[i1.z], x13 = x[i1.w];
            float s0 = fmaf(v0.x, x00, fmaf(v0.y, x01, fmaf(v0.z, x02, v0.w * x03)));
            float s1 = fmaf(v1.x, x10, fmaf(v1.y, x11, fmaf(v1.z, x12, v1.w * x13)));
            sum += s0 + s1;
        }
        // Tail: remaining single chunks.
        for (; c < nvec; c += 32) {
            vfloat4 v = __builtin_nontemporal_load(vv + c);
            vint4   i = __builtin_nontemporal_load(cv + c);
            sum = fmaf(v.x, x[i.x], sum);
            sum = fmaf(v.y, x[i.y], sum);
            sum = fmaf(v.z, x[i.z], sum);
            sum = fmaf(v.w, x[i.w], sum);
        }
        // Scalar remainder (len % 4 elements; none for this dataset).
        for (int k = start + (nvec << 2) + lane; k < end; k += 32) {
            sum = fmaf(__builtin_nontemporal_load(values + k), x[cols[k]], sum);
        }
    } else {
        // Unaligned fallback: scalar but still coalesced across the wave.
        for (int k = start + lane; k < end; k += 32) {
            sum = fmaf(__builtin_nontemporal_load(values + k), x[cols[k]], sum);
        }
    }

    // Wave32 tree reduction (gfx1250 is wave32-only; never assume 64 lanes).
    #pragma unroll
    for (int off = 16; off > 0; off >>= 1) {
        sum += __shfl_down(sum, off, 32);
    }

    if (lane == 0) {
        // y is written once and never re-read here -> NT store.
        __builtin_nontemporal_store(sum, y + row);
    }
}

extern "C" void kernel_launch(void* const* d_in, const int* in_sizes, int n_in,
                              void* d_out, int out_size, void* d_ws, size_t ws_size,
                              hipStream_t stream) {
    // setup_inputs() order: values (f32), col_indices (i32), row_ptrs (i32), x (f32)
    const float* values   = (const float*)d_in[0];
    const int*   cols     = (const int*)d_in[1];
    const int*   row_ptrs = (const int*)d_in[2];
    const float* x        = (const float*)d_in[3];
    float*       y        = (float*)d_out;

    const int num_rows = in_sizes[2] - 1;   // row_ptrs has num_rows+1 entries

    const int waves_per_block = 8;          // 256 threads = 8 wave32s
    dim3 block(32 * waves_per_block);
    dim3 grid((num_rows + waves_per_block - 1) / waves_per_block);

    spmv_csr_wave32_kernel<<<grid, block, 0, stream>>>(values, cols, row_ptrs, x, y, num_rows);
}